// GraphCutConditionalGainLoss_58411555225930
// MI455X (gfx1250) — compile-verified
//
#include <hip/hip_runtime.h>

#define NB 512
#define TPB 256
#define WPB (TPB / 32)              // 8 wave32s per block
#define TOTAL_WAVES (NB * WPB)      // 4096 waves
#define D 256
#define LAMDA 0.5f
#define ETA 1.0f
#define EPS 1e-8f

typedef __attribute__((ext_vector_type(2))) float v2f;
typedef __attribute__((ext_vector_type(8))) float v8f;

// ---------------------------------------------------------------------------
// Kernel 1: streaming pass over X. Each wave owns rows r = gw, gw+4096, ...
// Row layout: 32 lanes x 8 floats (two float4 / b128 loads per lane) = 256 = D.
// Per row: wave-reduced squared norm -> rinv, then 3 masked FMA accumulations.
// Deterministic cross-wave LDS reduce -> per-block partials (3 x 256 floats).
// ---------------------------------------------------------------------------
__global__ __launch_bounds__(TPB) void gc_partials(
    const float* __restrict__ X,
    const unsigned char* __restrict__ kn,
    const unsigned char* __restrict__ un,
    float* __restrict__ partials, int N) {
  __shared__ float lds[WPB * 3 * D];

  const int lane = threadIdx.x & 31;
  const int wib  = threadIdx.x >> 5;
  const int gw   = blockIdx.x * WPB + wib;

  float a0[8], a1[8], a2[8];
#pragma unroll
  for (int j = 0; j < 8; ++j) { a0[j] = 0.f; a1[j] = 0.f; a2[j] = 0.f; }

  for (int r = gw; r < N; r += TOTAL_WAVES) {
    const float4* row = (const float4*)(X + (size_t)r * D);
    // gfx1250 path: speculative prefetch of this wave's next row
    int rn = r + TOTAL_WAVES;
    if (rn < N) __builtin_prefetch(X + (size_t)rn * D + lane * 8, 0, 0);

    float4 va = row[lane * 2];
    float4 vb = row[lane * 2 + 1];

    float ss = va.x * va.x + va.y * va.y + va.z * va.z + va.w * va.w +
               vb.x * vb.x + vb.y * vb.y + vb.z * vb.z + vb.w * vb.w;
#pragma unroll
    for (int off = 16; off >= 1; off >>= 1) ss += __shfl_xor(ss, off, 32);

    float rinv = 1.0f / fmaxf(sqrtf(ss), EPS);
    float wk = kn[r] ? rinv : 0.0f;
    float wu = un[r] ? rinv : 0.0f;

    float v[8] = {va.x, va.y, va.z, va.w, vb.x, vb.y, vb.z, vb.w};
#pragma unroll
    for (int j = 0; j < 8; ++j) {
      a0[j] = fmaf(v[j], rinv, a0[j]);
      a1[j] = fmaf(v[j], wk,   a1[j]);
      a2[j] = fmaf(v[j], wu,   a2[j]);
    }
  }

  // each wave dumps its 3 partial vectors (lane l owns columns 8l..8l+7)
#pragma unroll
  for (int j = 0; j < 8; ++j) {
    lds[(wib * 3 + 0) * D + lane * 8 + j] = a0[j];
    lds[(wib * 3 + 1) * D + lane * 8 + j] = a1[j];
    lds[(wib * 3 + 2) * D + lane * 8 + j] = a2[j];
  }
  __syncthreads();

  const int t = threadIdx.x;
#pragma unroll
  for (int v3 = 0; v3 < 3; ++v3) {
    float s = 0.f;
#pragma unroll
    for (int w = 0; w < WPB; ++w) s += lds[(w * 3 + v3) * D + t];
    partials[((size_t)blockIdx.x * 3 + v3) * D + t] = s;
  }
}

// ---------------------------------------------------------------------------
// Kernel 2: reduce block partials (fixed order -> deterministic), then compute
// the 3x3 Gram matrix G = S^T S with V_WMMA_F32_16X16X4_F32 (exact f32 path).
// S is staged as a 4-row LDS table {s_all, s_known, s_unknown, zeros}; each
// lane resolves its row pointer ONCE, so the K-loop is branch-free:
// one ds_load_b64 + one wmma per step. A and B operands are identical here
// (symmetric Gram product, lane&15 indexes both M-row and N-col).
//   G[0][1] = s_all.s_known, G[1][1] = s_known.s_known, G[1][2] = s_known.s_unknown
// ---------------------------------------------------------------------------
__global__ __launch_bounds__(TPB) void gc_finalize(
    const float* __restrict__ partials, float* __restrict__ out, int N) {
  __shared__ float sMat[4][D];   // rows: s_all, s_known, s_unknown, zeros
  __shared__ float gmat[16][16];

  const int t = threadIdx.x;
  float r0 = 0.f, r1 = 0.f, r2 = 0.f;
  for (int b = 0; b < NB; ++b) {
    r0 += partials[((size_t)b * 3 + 0) * D + t];
    r1 += partials[((size_t)b * 3 + 1) * D + t];
    r2 += partials[((size_t)b * 3 + 2) * D + t];
  }
  sMat[0][t] = r0; sMat[1][t] = r1; sMat[2][t] = r2; sMat[3][t] = 0.0f;
  __syncthreads();

  if (threadIdx.x < 32) {              // wave 0 only; EXEC all-1s inside
    const int lane = threadIdx.x;
    const int m  = lane & 15;          // M row (A) / N col (B)
    const int hi = lane >> 4;          // half-wave selects K pair {0,1} vs {2,3}
    const float* rowp = &sMat[(m < 3) ? m : 3][2 * hi];  // lane-fixed operand row
    v8f c = {};
#pragma unroll 4
    for (int k0 = 0; k0 < D; k0 += 4) {
      // A lane layout (16x4 f32): VGPR0 = A[m][hi?2:0], VGPR1 = A[m][hi?3:1]
      v2f a = *(const v2f*)(rowp + k0);     // 8B-aligned ds_load_b64
      c = __builtin_amdgcn_wmma_f32_16x16x4_f32(
          /*neg_a=*/false, a, /*neg_b=*/false, a,
          /*c_mod=*/(short)0, c, /*reuse_a=*/false, /*reuse_b=*/false);
    }
    // C/D layout: VGPR j, lane -> element [j + 8*hi][lane&15]
#pragma unroll
    for (int j = 0; j < 8; ++j) gmat[hi * 8 + j][m] = c[j];
  }
  __syncthreads();

  if (t == 0) {
    float term1   = gmat[0][1];                      // dot(s_all, s_known)
    float term2   = LAMDA * gmat[1][1];              // lamda * dot(s_known, s_known)
    float penalty = 2.0f * LAMDA * ETA * gmat[1][2]; // 2*lamda*eta * dot(s_known, s_unknown)
    out[0] = -(term1 - term2 - penalty) / (float)N;
  }
}

extern "C" void kernel_launch(void* const* d_in, const int* in_sizes, int n_in,
                              void* d_out, int out_size, void* d_ws, size_t ws_size,
                              hipStream_t stream) {
  const float* X = (const float*)d_in[0];
  const unsigned char* kn = (const unsigned char*)d_in[1];  // jax bool = 1 byte
  const unsigned char* un = (const unsigned char*)d_in[2];
  const int N = in_sizes[1];                                // 200000 mask entries
  float* partials = (float*)d_ws;                           // NB * 3 * 256 floats

  gc_partials<<<NB, TPB, 0, stream>>>(X, kn, un, partials, N);
  gc_finalize<<<1, TPB, 0, stream>>>(partials, (float*)d_out, N);
}